// RNNDecoder_59751585021965
// MI455X (gfx1250) — compile-verified
//
#include <hip/hip_runtime.h>
#include <hip/hip_bf16.h>

// ---------------------------------------------------------------------------
// Pointer-generator RNN decoder single step for MI455X (gfx1250, wave32).
// All GEMMs use V_WMMA_F32_16X16X4_F32 (fp32 WMMA): inputs are fp32, the
// dominant vd2 GEMM is HBM-bound on its 102MB fp32 weight (a per-call bf16
// conversion pass would cost a full extra weight read+write), and the fused
// attention GEMM streams 52MB of activations once. The vd2 kernel overlaps
// the HBM weight stream with WMMA via GLOBAL_LOAD_ASYNC_TO_LDS_B128 double
// buffering (ASYNCcnt pipeline) in persistent workgroups.
// ---------------------------------------------------------------------------

typedef float v2f __attribute__((ext_vector_type(2)));
typedef float v8f __attribute__((ext_vector_type(8)));

#define B_   64
#define S_   400
#define V_   50000
#define E_   128
#define H_   512
#define M_   512
#define OOV_ 20
#define G3H_ (3 * H_)      // 1536
#define VO_  (V_ + OOV_)   // 50020
#define BS_  (B_ * S_)     // 25600

__device__ __forceinline__ v8f wmma_f32_k4(v2f a, v2f b, v8f c) {
  // D = A(16x4) * B(4x16) + C(16x16), fp32, wave32
  return __builtin_amdgcn_wmma_f32_16x16x4_f32(
      /*neg_a=*/false, a, /*neg_b=*/false, b,
      /*c_mod=*/(short)0, c, /*reuse_a=*/false, /*reuse_b=*/false);
}

// Generic -> LDS address-space cast to obtain the raw 32-bit LDS byte offset
// needed by GLOBAL_LOAD_ASYNC_TO_LDS (its VDST operand is an LDS address).
typedef __attribute__((address_space(3))) float lds_f;
__device__ __forceinline__ unsigned lds_offset_of(float* p) {
  return (unsigned)(uintptr_t)(lds_f*)p;
}

// Async global->LDS 128-bit load (GV mode), tracked by ASYNCcnt.
__device__ __forceinline__ void async_load_b128(unsigned lds_addr,
                                                const float* gaddr) {
  asm volatile("global_load_async_to_lds_b128 %0, %1, off"
               :: "v"(lds_addr), "v"(gaddr)
               : "memory");
}
__device__ __forceinline__ void wait_async_le4() {
  asm volatile("s_wait_asynccnt 4" ::: "memory");
}
__device__ __forceinline__ void wait_async_le0() {
  asm volatile("s_wait_asynccnt 0" ::: "memory");
}

// ---------------- reductions (deterministic trees) --------------------------
__device__ __forceinline__ float warpSum(float v) {
#pragma unroll
  for (int off = 16; off > 0; off >>= 1) v += __shfl_xor(v, off, 32);
  return v;
}
__device__ __forceinline__ float warpMax(float v) {
#pragma unroll
  for (int off = 16; off > 0; off >>= 1) v = fmaxf(v, __shfl_xor(v, off, 32));
  return v;
}
// nw = number of waves in the block (<= 16). swr must have >= nw floats.
__device__ __forceinline__ float blockSum(float v, float* swr, int tid, int nw) {
  int lane = tid & 31, wid = tid >> 5;
  v = warpSum(v);
  __syncthreads();
  if (lane == 0) swr[wid] = v;
  __syncthreads();
  float t = (tid < nw) ? swr[tid] : 0.f;
  if (wid == 0) t = warpSum(t);
  if (tid == 0) swr[0] = t;
  __syncthreads();
  return swr[0];
}
__device__ __forceinline__ float blockMax(float v, float* swr, int tid, int nw) {
  int lane = tid & 31, wid = tid >> 5;
  v = warpMax(v);
  __syncthreads();
  if (lane == 0) swr[wid] = v;
  __syncthreads();
  float t = (tid < nw) ? swr[tid] : -3.0e38f;
  if (wid == 0) t = warpMax(t);
  if (tid == 0) swr[0] = t;
  __syncthreads();
  return swr[0];
}

// ---------------- 1: embedding gather ---------------------------------------
__global__ void embed_kernel(const float* __restrict__ embW,
                             const int* __restrict__ y,
                             float* __restrict__ y_emb) {
  int idx = blockIdx.x * blockDim.x + threadIdx.x;
  if (idx >= B_ * E_) return;
  int b = idx >> 7, e = idx & (E_ - 1);
  y_emb[idx] = embW[(size_t)y[b] * E_ + e];
}

// ---------------- generic C = A[M,K] @ W[N,K]^T + bias, fp32 WMMA -----------
// One wave per 16x16 output tile. grid = (N/16, M/16), block = 32.
__global__ __launch_bounds__(32) void gemm_awt_kernel(
    const float* __restrict__ A, const float* __restrict__ W,
    const float* __restrict__ bias, float* __restrict__ C,
    int K, int ldc) {
  const int col0 = blockIdx.x * 16;
  const int row0 = blockIdx.y * 16;
  const int lane = threadIdx.x & 31;
  const int m  = lane & 15;
  const int ko = (lane >> 4) * 2;
  const float* __restrict__ Arow = A + (size_t)(row0 + m) * K;
  const float* __restrict__ Wrow = W + (size_t)(col0 + m) * K;
  v8f c = {};
  for (int k = 0; k < K; k += 4) {
    v2f a, b;
    a.x = Arow[k + ko]; a.y = Arow[k + ko + 1];
    b.x = Wrow[k + ko]; b.y = Wrow[k + ko + 1];
    c = wmma_f32_k4(a, b, c);
  }
  const int n = col0 + m;
  const float bn = bias ? bias[n] : 0.f;
  const int rbase = (lane >> 4) * 8;
#pragma unroll
  for (int r = 0; r < 8; ++r)
    C[(size_t)(row0 + rbase + r) * ldc + n] = c[r] + bn;
}

// ---------------- 2: GRU gates ---------------------------------------------
__global__ void gru_gate_kernel(const float* __restrict__ gi,
                                const float* __restrict__ gh,
                                const float* __restrict__ h0,
                                float* __restrict__ h1) {
  int idx = blockIdx.x * blockDim.x + threadIdx.x;
  if (idx >= B_ * H_) return;
  int b = idx >> 9, hh = idx & (H_ - 1);
  const float* gib = gi + (size_t)b * G3H_;
  const float* ghb = gh + (size_t)b * G3H_;
  float ir = gib[hh], iz = gib[H_ + hh], in_ = gib[2 * H_ + hh];
  float hr = ghb[hh], hz = ghb[H_ + hh], hn  = ghb[2 * H_ + hh];
  float r = 1.f / (1.f + __expf(-(ir + hr)));
  float z = 1.f / (1.f + __expf(-(iz + hz)));
  float n = tanhf(in_ + r * hn);
  h1[idx] = (1.f - z) * n + z * h0[idx];
}

// ---------------- 3: fused attention scores ---------------------------------
// scores[b,s] = masked( sum_h tanh(mb[b,s,:]·Wm[h,:] + dec[b,h] + cov[b,s]*Wc[h]) * v[h] )
// One WG of 4 waves handles 16 (b,s)-rows x all 512 h; each wave does 8
// 16-col tiles, K=512 in steps of 4 with fp32 WMMA. Deterministic reduce.
__global__ __launch_bounds__(128) void attn_scores_kernel(
    const float* __restrict__ mb, const float* __restrict__ Wm,
    const float* __restrict__ dec_feat, const float* __restrict__ Wc,
    const float* __restrict__ vvec, const float* __restrict__ coverage,
    const float* __restrict__ src_mask, float* __restrict__ scores) {
  __shared__ float swr[4][16];
  const int row0 = blockIdx.x * 16;
  const int tid  = threadIdx.x;
  const int wave = tid >> 5;
  const int lane = tid & 31;
  const int m    = lane & 15;
  const int ko   = (lane >> 4) * 2;
  const int rbase = (lane >> 4) * 8;
  const float* __restrict__ Arow = mb + (size_t)(row0 + m) * M_;

  float acc[8] = {0.f, 0.f, 0.f, 0.f, 0.f, 0.f, 0.f, 0.f};
  for (int t = 0; t < 8; ++t) {
    const int col0 = (wave * 8 + t) * 16;
    const float* __restrict__ Wrow = Wm + (size_t)(col0 + m) * M_;
    v8f c = {};
    for (int k = 0; k < M_; k += 4) {
      v2f a, b;
      a.x = Arow[k + ko]; a.y = Arow[k + ko + 1];
      b.x = Wrow[k + ko]; b.y = Wrow[k + ko + 1];
      c = wmma_f32_k4(a, b, c);
    }
    const int h = col0 + m;         // this lane's output column
    const float vh  = vvec[h];
    const float wch = Wc[h];
#pragma unroll
    for (int r = 0; r < 8; ++r) {
      const int row  = row0 + rbase + r;
      const int bidx = row / S_;    // tiles never straddle b (S % 16 == 0)
      float val = c[r] + dec_feat[(size_t)bidx * H_ + h] + coverage[row] * wch;
      acc[r] += tanhf(val) * vh;
    }
  }
  // reduce over the 16 h-columns inside each half-wave (fixed tree)
#pragma unroll
  for (int off = 1; off < 16; off <<= 1)
#pragma unroll
    for (int r = 0; r < 8; ++r) acc[r] += __shfl_xor(acc[r], off, 32);
  if (m == 0) {
#pragma unroll
    for (int r = 0; r < 8; ++r) swr[wave][rbase + r] = acc[r];
  }
  __syncthreads();
  if (tid < 16) {
    const int row = row0 + tid;
    float s = ((swr[0][tid] + swr[1][tid]) + (swr[2][tid] + swr[3][tid]));
    scores[row] = (src_mask[row] > 0.f) ? s : -1e9f;
  }
}

// ---------------- 4: masked softmax + coverage + context + pack -------------
__global__ __launch_bounds__(512) void attn_softmax_ctx_kernel(
    const float* __restrict__ scores, const float* __restrict__ src_mask,
    const float* __restrict__ coverage, const float* __restrict__ mb,
    const float* __restrict__ h1,
    float* __restrict__ attn_out, float* __restrict__ cov_out,
    float* __restrict__ ctx_out, float* __restrict__ vocab_in) {
  __shared__ float sattn[S_];
  __shared__ float swr[16];
  const int b = blockIdx.x;
  const int tid = threadIdx.x;

  float x = (tid < S_) ? scores[(size_t)b * S_ + tid] : -3.0e38f;
  float mx = blockMax(x, swr, tid, 16);
  float e = (tid < S_) ? __expf(x - mx) : 0.f;
  float s1 = blockSum(e, swr, tid, 16);
  float araw = e / s1;
  float am = (tid < S_) ? araw * src_mask[(size_t)b * S_ + tid] : 0.f;
  float s2 = blockSum(am, swr, tid, 16);
  float a = am / (s2 + 1e-10f);
  if (tid < S_) {
    sattn[tid] = a;
    attn_out[(size_t)b * S_ + tid] = a;
    cov_out[(size_t)b * S_ + tid] = coverage[(size_t)b * S_ + tid] + a;
  }
  __syncthreads();
  // word_context[b, tid] = sum_s attn[s] * mb[b, s, tid]   (tid = 0..511)
  const float* __restrict__ mbb = mb + (size_t)b * S_ * M_;
  float acc = 0.f;
  for (int s = 0; s < S_; ++s) acc += sattn[s] * mbb[(size_t)s * M_ + tid];
  ctx_out[(size_t)b * M_ + tid] = acc;
  vocab_in[(size_t)b * (M_ + H_) + tid] = acc;          // [ctx | h1] for vd1
  vocab_in[(size_t)b * (M_ + H_) + M_ + tid] = h1[(size_t)b * H_ + tid];
}

// ---------------- 5: p_gen ---------------------------------------------------
__global__ __launch_bounds__(128) void pgen_kernel(
    const float* __restrict__ ctx, const float* __restrict__ h1,
    const float* __restrict__ y_emb, const float* __restrict__ W,
    const float* __restrict__ bptr, float* __restrict__ pgen_out) {
  __shared__ float swr[4];
  const int b = blockIdx.x, tid = threadIdx.x;
  float acc = 0.f;
  for (int k = tid; k < M_ + H_ + E_; k += 128) {
    float x = (k < M_) ? ctx[(size_t)b * M_ + k]
            : (k < M_ + H_) ? h1[(size_t)b * H_ + (k - M_)]
                            : y_emb[(size_t)b * E_ + (k - M_ - H_)];
    acc += x * W[k];
  }
  float s = blockSum(acc, swr, tid, 4);
  if (tid == 0) pgen_out[b] = 1.f / (1.f + __expf(-(s + bptr[0])));
}

// ---------------- 6: logits = hid[64,512] @ vd2_W[V,512]^T + b --------------
// HBM-bound (102MB weight). Persistent WGs: 625 blocks x 5 column tiles each;
// each 16-row x 128-col weight K-panel streams into LDS with
// GLOBAL_LOAD_ASYNC_TO_LDS_B128, double-buffered so the next panel's HBM
// traffic overlaps the current panel's WMMAs. LDS row stride 132 floats:
// 528B rows (16B-aligned for b128 async writes) and 132%64==4 -> the 16
// B-matrix lanes hit 16 distinct banks (conflict-free ds reads).
#define VD2_NWG 625
#define VD2_TPW 5                   // column tiles per workgroup (625*5=3125)
#define VD2_LDSTRIDE 132            // floats per LDS panel row
#define VD2_PANEL 128               // K-panel width in floats (4 panels of 512)

__device__ __forceinline__ void vd2_issue_panel(const float* __restrict__ W,
                                                int col0, int kp,
                                                unsigned lds_base, int tid) {
#pragma unroll
  for (int i = 0; i < 4; ++i) {
    const int c  = i * 128 + tid;    // 512 x 16B chunks per panel
    const int n  = c >> 5;           // row within the 16-wide column tile
    const int kk = (c & 31) << 2;    // float offset within panel row
    const float* g = W + (size_t)(col0 + n) * H_ + kp * VD2_PANEL + kk;
    const unsigned l = lds_base + (unsigned)((n * VD2_LDSTRIDE + kk) * 4);
    async_load_b128(l, g);
  }
}

__global__ __launch_bounds__(128) void vd2_async_kernel(
    const float* __restrict__ hid, const float* __restrict__ W,
    const float* __restrict__ bias, float* __restrict__ logits) {
  __shared__ float bsm[2][16 * VD2_LDSTRIDE];
  const int tid  = threadIdx.x;
  const int wave = tid >> 5;
  const int lane = tid & 31;
  const int m     = lane & 15;
  const int ko    = (lane >> 4) * 2;
  const int rbase = (lane >> 4) * 8;
  const int row0  = wave * 16;
  const int tile0 = blockIdx.x * VD2_TPW;
  const unsigned lds_base[2] = {lds_offset_of(&bsm[0][0]),
                                lds_offset_of(&bsm[1][0])};
  const float* __restrict__ Arow = hid + (size_t)(row0 + m) * H_;
  const int NSTEPS = VD2_TPW * 4;   // (tile, panel) steps

  vd2_issue_panel(W, tile0 * 16, 0, lds_base[0], tid);  // prologue
  v8f c = {};
  for (int s = 0; s < NSTEPS; ++s) {
    if (s + 1 < NSTEPS) {
      const int t2 = (s + 1) >> 2, p2 = (s + 1) & 3;
      vd2_issue_panel(W, (tile0 + t2) * 16, p2, lds_base[(s + 1) & 1], tid);
      wait_async_le4();             // 4 newest in flight; current panel landed
    } else {
      wait_async_le0();
    }
    __syncthreads();
    const float* __restrict__ Brow = &bsm[s & 1][m * VD2_LDSTRIDE];
    const int kbase = (s & 3) * VD2_PANEL;
#pragma unroll 8
    for (int k = 0; k < VD2_PANEL; k += 4) {
      v2f a, b;
      a.x = Arow[kbase + k + ko]; a.y = Arow[kbase + k + ko + 1];
      b.x = Brow[k + ko];         b.y = Brow[k + ko + 1];
      c = wmma_f32_k4(a, b, c);
    }
    if ((s & 3) == 3) {             // last K-panel of this column tile
      const int n = (tile0 + (s >> 2)) * 16 + m;
      const float bn = bias[n];
#pragma unroll
      for (int r = 0; r < 8; ++r) {
        logits[(size_t)(row0 + rbase + r) * V_ + n] = c[r] + bn;
        c[r] = 0.f;
      }
    }
    __syncthreads();                // readers done before buffer is re-filled
  }
}

// ---------------- 7: per-row max & sum(exp) over V ---------------------------
__global__ __launch_bounds__(256) void rowstat_kernel(
    const float* __restrict__ logits, float* __restrict__ rmax,
    float* __restrict__ rsum) {
  __shared__ float swr[8];
  const int b = blockIdx.x, tid = threadIdx.x;
  const float* __restrict__ row = logits + (size_t)b * V_;
  float m = -3.0e38f;
  for (int k = tid; k < V_; k += 256) m = fmaxf(m, row[k]);
  m = blockMax(m, swr, tid, 8);
  float s = 0.f;
  for (int k = tid; k < V_; k += 256) s += __expf(row[k] - m);
  s = blockSum(s, swr, tid, 8);
  if (tid == 0) { rmax[b] = m; rsum[b] = s; }
}

// ---------------- 8: base final distribution --------------------------------
__global__ void final_base_kernel(const float* __restrict__ logits,
                                  const float* __restrict__ rmax,
                                  const float* __restrict__ rsum,
                                  const float* __restrict__ pgen,
                                  float* __restrict__ outf) {
  size_t idx = (size_t)blockIdx.x * blockDim.x + threadIdx.x;
  if (idx >= (size_t)B_ * VO_) return;
  int b = (int)(idx / VO_);
  int n = (int)(idx % VO_);
  float v = 0.f;
  if (n < V_)
    v = pgen[b] * __expf(logits[(size_t)b * V_ + n] - rmax[b]) / rsum[b];
  outf[idx] = v;
}

// ---------------- 9: deterministic scatter-add (one thread per row) ---------
__global__ void scatter_kernel(const int* __restrict__ src_oov,
                               const float* __restrict__ attn,
                               const float* __restrict__ pgen,
                               float* __restrict__ outf) {
  int b = blockIdx.x * blockDim.x + threadIdx.x;
  if (b >= B_) return;
  const float g = 1.f - pgen[b];
  float* row = outf + (size_t)b * VO_;
  for (int s = 0; s < S_; ++s)
    row[src_oov[(size_t)b * S_ + s]] += g * attn[(size_t)b * S_ + s];
}

// ---------------------------------------------------------------------------
extern "C" void kernel_launch(void* const* d_in, const int* in_sizes, int n_in,
                              void* d_out, int out_size, void* d_ws,
                              size_t ws_size, hipStream_t stream) {
  (void)in_sizes; (void)n_in; (void)out_size; (void)ws_size;
  // ---- inputs (setup_inputs order) ----
  const int*   y        = (const int*)  d_in[0];
  const float* h0       = (const float*)d_in[1];   // [1,B,H] == [B,H]
  const float* mb       = (const float*)d_in[2];   // [B,S,M]
  const float* src_mask = (const float*)d_in[3];
  /* d_in[4] = max_num_oovs (compile-time OOV_) */
  const int*   src_oov  = (const int*)  d_in[5];
  const float* coverage = (const float*)d_in[6];
  const float* emb_W    = (const float*)d_in[7];
  const float* gru_Wih  = (const float*)d_in[8];
  const float* gru_Whh  = (const float*)d_in[9];
  const float* gru_bih  = (const float*)d_in[10];
  const float* gru_bhh  = (const float*)d_in[11];
  const float* attn_Wm  = (const float*)d_in[12];
  const float* attn_Wd  = (const float*)d_in[13];
  const float* attn_bd  = (const float*)d_in[14];
  const float* attn_Wc  = (const float*)d_in[15];  // [H,1]
  const float* attn_v   = (const float*)d_in[16];
  const float* pgen_W   = (const float*)d_in[17];  // [1, E+H+M]
  const float* pgen_b   = (const float*)d_in[18];
  const float* vd1_W    = (const float*)d_in[19];  // [H, H+M]
  const float* vd1_b    = (const float*)d_in[20];
  const float* vd2_W    = (const float*)d_in[21];  // [V, H]
  const float* vd2_b    = (const float*)d_in[22];

  // ---- outputs, concatenated flat in reference return order ----
  float* out       = (float*)d_out;
  float* out_final = out;                          // [B, V+OOV]
  float* out_h     = out_final + (size_t)B_ * VO_; // [1,B,H]
  float* out_ctx   = out_h + (size_t)B_ * H_;      // [B, M]
  float* out_attn  = out_ctx + (size_t)B_ * M_;    // [B, S]
  float* out_pgen  = out_attn + (size_t)B_ * S_;   // [B, 1]
  float* out_cov   = out_pgen + B_;                // [B, S]

  // ---- workspace layout (floats) ----
  float* ws       = (float*)d_ws;
  float* y_emb    = ws;                        // B*E          = 8192
  float* gi       = y_emb + B_ * E_;           // B*3H         = 98304
  float* gh       = gi + B_ * G3H_;            // B*3H         = 98304
  float* dec_feat = gh + B_ * G3H_;            // B*H          = 32768
  float* scores   = dec_feat + B_ * H_;        // B*S          = 25600
  float* vocab_in = scores + BS_;              // B*(M+H)      = 65536
  float* hid      = vocab_in + B_ * (M_ + H_); // B*H          = 32768
  float* rmax     = hid + B_ * H_;             // B
  float* rsum     = rmax + B_;                 // B
  float* logits   = rsum + B_;                 // B*V          = 3,200,000

  // 1. embedding gather
  embed_kernel<<<(B_ * E_ + 255) / 256, 256, 0, stream>>>(emb_W, y, y_emb);
  // 2. GRU input/hidden GEMMs (WMMA fp32)
  gemm_awt_kernel<<<dim3(G3H_ / 16, B_ / 16), 32, 0, stream>>>(
      y_emb, gru_Wih, gru_bih, gi, E_, G3H_);
  gemm_awt_kernel<<<dim3(G3H_ / 16, B_ / 16), 32, 0, stream>>>(
      h0, gru_Whh, gru_bhh, gh, H_, G3H_);
  // 3. GRU gate math -> h1 (written straight into h_next output)
  gru_gate_kernel<<<(B_ * H_ + 255) / 256, 256, 0, stream>>>(gi, gh, h0, out_h);
  // 4. dec_feat = h1 @ Wd^T + bd
  gemm_awt_kernel<<<dim3(H_ / 16, B_ / 16), 32, 0, stream>>>(
      out_h, attn_Wd, attn_bd, dec_feat, H_, H_);
  // 5. fused attention scores (dominant 13.4 GFLOP WMMA GEMM + tanh·v reduce)
  attn_scores_kernel<<<BS_ / 16, 128, 0, stream>>>(
      mb, attn_Wm, dec_feat, attn_Wc, attn_v, coverage, src_mask, scores);
  // 6. masked softmax, coverage update, context, pack [ctx|h1]
  attn_softmax_ctx_kernel<<<B_, 512, 0, stream>>>(
      scores, src_mask, coverage, mb, out_h, out_attn, out_cov, out_ctx,
      vocab_in);
  // 7. hid = [ctx|h1] @ vd1_W^T + b
  gemm_awt_kernel<<<dim3(H_ / 16, B_ / 16), 32, 0, stream>>>(
      vocab_in, vd1_W, vd1_b, hid, M_ + H_, H_);
  // 8. p_gen
  pgen_kernel<<<B_, 128, 0, stream>>>(out_ctx, out_h, y_emb, pgen_W, pgen_b,
                                      out_pgen);
  // 9. logits = hid @ vd2_W^T + b (async double-buffered weight stream + WMMA)
  vd2_async_kernel<<<VD2_NWG, 128, 0, stream>>>(hid, vd2_W, vd2_b, logits);
  // 10. per-row softmax stats over V
  rowstat_kernel<<<B_, 256, 0, stream>>>(logits, rmax, rsum);
  // 11. base final dist = p_gen * softmax(logits), zeros in OOV tail
  final_base_kernel<<<(int)(((size_t)B_ * VO_ + 255) / 256), 256, 0, stream>>>(
      logits, rmax, rsum, out_pgen, out_final);
  // 12. deterministic scatter-add of (1-p_gen)*attn via src_oov
  scatter_kernel<<<(B_ + 63) / 64, 64, 0, stream>>>(src_oov, out_attn,
                                                    out_pgen, out_final);
}